// GraphSAGE_34402688041309
// MI455X (gfx1250) — compile-verified
//
#include <hip/hip_runtime.h>

typedef __attribute__((ext_vector_type(2))) float v2f;
typedef __attribute__((ext_vector_type(4))) float v4f;
typedef __attribute__((ext_vector_type(8))) float v8f;

#define N_NODES 100000
#define N_EDGES 1600000
#define CH      128
#define SPITCH  132   // LDS row pitch: row*132 % 64 banks advances 4/row -> conflict-free frag reads

// ---------------------------------------------------------------------------
// Degree count (int atomics in L2)
// ---------------------------------------------------------------------------
__global__ __launch_bounds__(256) void deg_kernel(const int* __restrict__ dst,
                                                  int* __restrict__ cnt) {
    int e = blockIdx.x * 256 + threadIdx.x;
    if (e < N_EDGES) atomicAdd(&cnt[dst[e]], 1);
}

__global__ __launch_bounds__(256) void inv_kernel(const int* __restrict__ cnt,
                                                  float* __restrict__ inv) {
    int i = blockIdx.x * 256 + threadIdx.x;
    if (i < N_NODES) {
        int c = cnt[i];
        inv[i] = 1.0f / (float)(c > 1 ? c : 1);
    }
}

// ---------------------------------------------------------------------------
// Edge scatter: msg[dst] += h[src].  One edge per half-block, one channel per
// lane -> coalesced 512B gather per edge, fp32 hardware atomics into L2-resident
// accumulator (51MB << 192MB L2).
// ---------------------------------------------------------------------------
__global__ __launch_bounds__(256) void scatter_kernel(const float* __restrict__ h,
                                                      const int* __restrict__ src,
                                                      const int* __restrict__ dst,
                                                      float* __restrict__ msg) {
    int e = blockIdx.x * 2 + (threadIdx.x >> 7);
    int c = threadIdx.x & 127;
    int s = src[e];
    int d = dst[e];
    unsafeAtomicAdd(&msg[(size_t)d * CH + c], h[(size_t)s * CH + c]);
}

// ---------------------------------------------------------------------------
// Fused SAGEConv GEMM: out = relu( (msg*inv) @ Wl^T + bl + root @ Wr^T )
// One block = 16-node M-tile; 8 wave32, one 16-col N-tile each.
// f32 WMMA 16x16x4, K=128 per weight matrix -> 64 v_wmma per wave.
// ---------------------------------------------------------------------------
__global__ __launch_bounds__(256)
void sage_gemm_kernel(const float* __restrict__ msg, const float* __restrict__ root,
                      const float* __restrict__ inv,
                      const float* __restrict__ Wl, const float* __restrict__ bl,
                      const float* __restrict__ Wr, float* __restrict__ out) {
    __shared__ float sMean[16 * SPITCH];
    __shared__ float sRoot[16 * SPITCH];

    const int tid = threadIdx.x;
    const int m0  = blockIdx.x * 16;   // 100000 / 16 == 6250 exact: no partial tiles

    // Cooperative stage of the 16x128 activation tiles (coalesced global reads).
    for (int i = tid; i < 16 * CH; i += 256) {
        int r = i >> 7, c = i & 127;
        size_t row = (size_t)(m0 + r);
        sMean[r * SPITCH + c] = msg[row * CH + c] * inv[row];
        sRoot[r * SPITCH + c] = root[row * CH + c];
    }
    __syncthreads();

    const int lane  = tid & 31;
    const int wave  = tid >> 5;
    const int n0    = wave * 16;
    const int rlo   = lane & 15;            // M for A-frag / N for B,C frags
    const int khalf = (lane >> 4) << 1;     // K offset 0 or 2 per half-wave

    // C/D layout: acc[r] holds C[M = r + 8*(lane>=16)][N = n0 + rlo].
    // Bias depends only on N -> broadcast into all 8 slots.
    v8f acc;
    {
        float b = bl[n0 + rlo];
#pragma unroll
        for (int r = 0; r < 8; ++r) acc[r] = b;
    }

    const float* aMean = sMean + rlo * SPITCH;                 // A: row = lane&15
    const float* aRoot = sRoot + rlo * SPITCH;
    const float* wlRow = Wl + (size_t)(n0 + rlo) * CH;         // B: W^T, col = lane&15
    const float* wrRow = Wr + (size_t)(n0 + rlo) * CH;

    // mean @ Wl^T
#pragma unroll 8
    for (int k0 = 0; k0 < CH; k0 += 4) {
        v2f a = *(const v2f*)(aMean + k0 + khalf);   // ds_load_b64, conflict-free
        v2f b = *(const v2f*)(wlRow + k0 + khalf);   // global_load_b64, L2-resident
        acc = __builtin_amdgcn_wmma_f32_16x16x4_f32(false, a, false, b,
                                                    (short)0, acc, false, false);
    }
    // + root @ Wr^T
#pragma unroll 8
    for (int k0 = 0; k0 < CH; k0 += 4) {
        v2f a = *(const v2f*)(aRoot + k0 + khalf);
        v2f b = *(const v2f*)(wrRow + k0 + khalf);
        acc = __builtin_amdgcn_wmma_f32_16x16x4_f32(false, a, false, b,
                                                    (short)0, acc, false, false);
    }

    // ReLU + store (in-place over `root` is safe for layer 1: this block already
    // consumed its 16 rows into LDS/registers).
    const int col   = n0 + rlo;
    const int rbase = m0 + ((lane >> 4) << 3);
#pragma unroll
    for (int r = 0; r < 8; ++r) {
        float v = acc[r] > 0.0f ? acc[r] : 0.0f;
        out[(size_t)(rbase + r) * CH + col] = v;
    }
}

// ---------------------------------------------------------------------------
// Final head: out[n, 0:2] = h[n] @ fc_w^T + fc_b.  Wave per node, float4 loads,
// shuffle reduction across 32 lanes.
// ---------------------------------------------------------------------------
__global__ __launch_bounds__(256)
void head_kernel(const float* __restrict__ h, const float* __restrict__ fw,
                 const float* __restrict__ fb, float* __restrict__ out) {
    int lane = threadIdx.x & 31;
    int node = blockIdx.x * 8 + (threadIdx.x >> 5);
    if (node >= N_NODES) return;

    v4f hv = *(const v4f*)(h + (size_t)node * CH + lane * 4);
    v4f w0 = *(const v4f*)(fw + 0 * CH + lane * 4);
    v4f w1 = *(const v4f*)(fw + 1 * CH + lane * 4);

    float a0 = hv.x * w0.x + hv.y * w0.y + hv.z * w0.z + hv.w * w0.w;
    float a1 = hv.x * w1.x + hv.y * w1.y + hv.z * w1.z + hv.w * w1.w;
#pragma unroll
    for (int off = 16; off > 0; off >>= 1) {
        a0 += __shfl_xor(a0, off, 32);
        a1 += __shfl_xor(a1, off, 32);
    }
    if (lane == 0) {
        out[(size_t)node * 2 + 0] = a0 + fb[0];
        out[(size_t)node * 2 + 1] = a1 + fb[1];
    }
}

// ---------------------------------------------------------------------------
extern "C" void kernel_launch(void* const* d_in, const int* in_sizes, int n_in,
                              void* d_out, int out_size, void* d_ws, size_t ws_size,
                              hipStream_t stream) {
    const float* x   = (const float*)d_in[0];
    const int*   ei  = (const int*)d_in[1];
    const float* Wl0 = (const float*)d_in[2];
    const float* bl0 = (const float*)d_in[3];
    const float* Wr0 = (const float*)d_in[4];
    const float* Wl1 = (const float*)d_in[5];
    const float* bl1 = (const float*)d_in[6];
    const float* Wr1 = (const float*)d_in[7];
    const float* fw  = (const float*)d_in[8];
    const float* fb  = (const float*)d_in[9];
    float*       out = (float*)d_out;

    const int* src = ei;             // edge_index[0, :]
    const int* dst = ei + N_EDGES;   // edge_index[1, :]

    // Workspace layout: msg[N*CH] | h1[N*CH] | inv[N] | cnt[N]  (~103 MB)
    float* msg  = (float*)d_ws;
    float* h1   = msg + (size_t)N_NODES * CH;
    float* invd = h1 + (size_t)N_NODES * CH;
    int*   cnt  = (int*)(invd + N_NODES);

    const size_t featBytes = sizeof(float) * (size_t)N_NODES * CH;

    // Degrees (same for both layers).
    hipMemsetAsync(cnt, 0, sizeof(int) * N_NODES, stream);
    deg_kernel<<<(N_EDGES + 255) / 256, 256, 0, stream>>>(dst, cnt);
    inv_kernel<<<(N_NODES + 255) / 256, 256, 0, stream>>>(cnt, invd);

    // Layer 0
    hipMemsetAsync(msg, 0, featBytes, stream);
    scatter_kernel<<<N_EDGES / 2, 256, 0, stream>>>(x, src, dst, msg);
    sage_gemm_kernel<<<N_NODES / 16, 256, 0, stream>>>(msg, x, invd, Wl0, bl0, Wr0, h1);

    // Layer 1 (in-place output over h1)
    hipMemsetAsync(msg, 0, featBytes, stream);
    scatter_kernel<<<N_EDGES / 2, 256, 0, stream>>>(h1, src, dst, msg);
    sage_gemm_kernel<<<N_NODES / 16, 256, 0, stream>>>(msg, h1, invd, Wl1, bl1, Wr1, h1);

    // Head
    head_kernel<<<N_NODES / 8, 256, 0, stream>>>(h1, fw, fb, out);
}